// ClusteringCLR_43920335569344
// MI455X (gfx1250) — compile-verified
//
#include <hip/hip_runtime.h>
#include <hip/hip_bf16.h>
#include <math.h>

#define N 1024
#define CDIM 8
#define NPOW 24          // subspace power iterations per eigensolve
#define NEWTON 20
#define EPSF 2.2204e-16f
#define KSLICES 8        // split-K factor for the 1024-deep GEMM

typedef __attribute__((ext_vector_type(2))) float    v2f;
typedef __attribute__((ext_vector_type(8))) float    v8f;
typedef __attribute__((ext_vector_type(4))) unsigned v4u;
typedef __attribute__((ext_vector_type(8))) int      v8i;
typedef __attribute__((ext_vector_type(4))) int      v4i;

// scal[] indices (device-side control state; deterministic per launch)
#define SC_LAM   0
#define SC_DONE  1
#define SC_USEF  2
#define SC_SIGMA 3

// ---------------------------------------------------------------------------
// TDM helpers
#if __has_include(<hip/amd_detail/amd_gfx1250_TDM.h>)
#define TDM_LOAD(g0, g1, g2, g3, cpol) \
  __builtin_amdgcn_tensor_load_to_lds(g0, g1, g2, g3, (v8i){0,0,0,0,0,0,0,0}, cpol)
#else
#define TDM_LOAD(g0, g1, g2, g3, cpol) \
  __builtin_amdgcn_tensor_load_to_lds(g0, g1, g2, g3, cpol)
#endif

typedef __attribute__((address_space(3))) void lds_void_t;

__device__ __forceinline__ unsigned lds_addr_of(const void* p) {
  return (unsigned)(uintptr_t)(lds_void_t*)(void*)p;
}

// DMA a [rows x 16] f32 tile (row stride 16 floats) from global into LDS.
// D# packing per CDNA5 ISA 8.3/8.4: group0 = {flags, lds_addr, global_addr, type},
// group1 = {mask/data_size, dims, tile dims, strides}; groups 2/3 zero (2D tile).
__device__ __forceinline__ void tdm_load_tile16(const float* gsrc,
                                                unsigned ldsByteAddr, int rows) {
  unsigned long long ga = (unsigned long long)(uintptr_t)gsrc;
  v4u g0;
  g0.x = 1u;                                        // count=1 (valid), user mode
  g0.y = ldsByteAddr;                               // lds_addr
  g0.z = (unsigned)(ga & 0xFFFFFFFFu);              // global_addr[31:0]
  g0.w = (unsigned)((ga >> 32) & 0x01FFFFFFu) | (2u << 30);  // addr[56:32] | type=2
  unsigned td0 = 16u;                               // tensor_dim0 (row width)
  unsigned td1 = (unsigned)rows;                    // tensor_dim1 (rows)
  v8i g1;
  g1[0] = (int)(2u << 16);                          // workgroup_mask=0, data_size=4B
  g1[1] = (int)((td0 & 0xFFFFu) << 16);             // [31:16]=tensor_dim0[15:0]
  g1[2] = (int)(((td0 >> 16) & 0xFFFFu) | ((td1 & 0xFFFFu) << 16));
  g1[3] = (int)(((td1 >> 16) & 0xFFFFu) | (16u << 16));   // tile_dim0 = 16
  g1[4] = (int)((unsigned)rows & 0xFFFFu);          // tile_dim1 = rows, tile_dim2 = 0
  g1[5] = 16;                                       // tensor_dim0_stride[31:0] = 16
  g1[6] = 0;                                        // stride hi / dim1_stride lo
  g1[7] = 0;
  v4i gz = {0, 0, 0, 0};
  TDM_LOAD(g0, g1, gz, gz, 0);
}

// ---------------------------------------------------------------------------
__global__ void k_init(float* scal) {
  if (threadIdx.x == 0) {
    scal[SC_LAM] = 0.01f;
    scal[SC_DONE] = 0.f;
    scal[SC_USEF] = 0.f;
    scal[SC_SIGMA] = 0.f;
  }
}

__global__ __launch_bounds__(256) void k_preprocess(const float* __restrict__ Ain,
                                                    float* __restrict__ A0,
                                                    float* __restrict__ Afin,
                                                    float* __restrict__ U) {
  size_t idx = (size_t)blockIdx.x * 256 + threadIdx.x;
  int i = (int)(idx >> 10), j = (int)(idx & (N - 1));
  float a  = (i == j) ? 0.f : Ain[idx];
  float at = (i == j) ? 0.f : Ain[(size_t)j * N + i];
  A0[idx] = a;
  Afin[idx] = 0.5f * (a + at);
  U[idx] = 1.f;
}

__global__ __launch_bounds__(256) void k_rowsum(const float* __restrict__ M,
                                                float* __restrict__ deg) {
  __shared__ float red[256];
  int i = blockIdx.x, t = threadIdx.x;
  float s = 0.f;
  for (int j = t; j < N; j += 256) s += M[(size_t)i * N + j];
  red[t] = s; __syncthreads();
  for (int off = 128; off > 0; off >>= 1) {
    if (t < off) red[t] += red[t + off];
    __syncthreads();
  }
  if (t == 0) deg[i] = red[0];
}

__global__ __launch_bounds__(256) void k_maxdeg(const float* __restrict__ deg,
                                                float* __restrict__ scal) {
  __shared__ float red[256];
  int t = threadIdx.x;
  float m = 0.f;
  for (int i = t; i < N; i += 256) m = fmaxf(m, deg[i]);
  red[t] = m; __syncthreads();
  for (int off = 128; off > 0; off >>= 1) {
    if (t < off) red[t] = fmaxf(red[t], red[t + off]);
    __syncthreads();
  }
  if (t == 0) scal[SC_SIGMA] = 2.f * red[0] + 1.f;
}

__global__ __launch_bounds__(256) void k_initX(float* __restrict__ X) {
  int idx = blockIdx.x * 256 + threadIdx.x;   // N*16
  unsigned h = (unsigned)idx * 2654435761u + 0x9e3779b9u;
  h ^= h >> 16; h *= 0x85ebca6bu; h ^= h >> 13;
  X[idx] = ((float)(h & 0xFFFFu) / 65536.0f) - 0.5f;
}

// ---------------------------------------------------------------------------
// C_part[M x 16] = A[M x kLen slice, row-major lda] @ B[kStart.. x 16]
// grid = (8, nslices); slice s handles K-range [s*kLen, (s+1)*kLen) and writes
// its partial product to C + s*N*16. One wave owns a 16-row stripe.
// B panel is DMA'd into double-buffered LDS by the Tensor Data Mover while the
// WMMA loop consumes the other buffer (s_wait_tensorcnt + barrier handshake).
__global__ __launch_bounds__(256) void k_gemm_wmma(const float* __restrict__ A, int lda,
                                                   const float* __restrict__ B,
                                                   float* __restrict__ C, int kLen) {
  __shared__ float Bs[2][64 * 16];
  const int lane  = threadIdx.x & 31;
  const int wave  = threadIdx.x >> 5;
  const int m0    = (blockIdx.x * 8 + wave) * 16;
  const int row   = m0 + (lane & 15);
  const int khalf = (lane >> 4) << 1;   // 0 or 2: K-pair this half-wave holds
  const int n     = lane & 15;
  const int kStart = blockIdx.y * kLen;
  float* Cp = C + (size_t)blockIdx.y * N * 16;

  // prologue: DMA first chunk
  const int chunk0 = (kLen < 64) ? kLen : 64;
  if (threadIdx.x < 32)
    tdm_load_tile16(B + (size_t)kStart * 16, lds_addr_of(&Bs[0][0]), chunk0);
  __builtin_amdgcn_s_wait_tensorcnt(0);
  __syncthreads();

  v8f acc = {};
  int buf = 0;
  for (int kb = 0; kb < kLen; kb += 64) {
    const int chunk = (kLen - kb < 64) ? (kLen - kb) : 64;
    const int nkb = kb + 64;
    if (nkb < kLen) {                 // prefetch next chunk into the other buffer
      const int nchunk = (kLen - nkb < 64) ? (kLen - nkb) : 64;
      if (threadIdx.x < 32)
        tdm_load_tile16(B + (size_t)(kStart + nkb) * 16,
                        lds_addr_of(&Bs[buf ^ 1][0]), nchunk);
      __builtin_prefetch((const void*)(A + (size_t)row * lda + kStart + nkb), 0, 1);
    }
    for (int k4 = 0; k4 < chunk; k4 += 16) {
#pragma unroll
      for (int kk = 0; kk < 16; kk += 4) {
        const int k = k4 + kk;
        v2f a, b;
        const float* ap = A + (size_t)row * lda + kStart + kb + k + khalf;
        a.x = ap[0];                  // A[M=lane&15][K=khalf]
        a.y = ap[1];                  // A[M=lane&15][K=khalf+1]
        const float* bp = &Bs[buf][(k + khalf) * 16 + n];
        b.x = bp[0];                  // B[K=khalf][N=n]
        b.y = bp[16];                 // B[K=khalf+1][N=n]
        acc = __builtin_amdgcn_wmma_f32_16x16x4_f32(
            false, a, false, b, (short)0, acc, false, false);
      }
    }
    __builtin_amdgcn_s_wait_tensorcnt(0);   // next chunk's DMA done (wave 0)
    __syncthreads();                        // visible to all waves
    buf ^= 1;
  }
  const int mbase = m0 + ((lane >> 4) << 3);   // lanes 16-31 hold M=8..15
#pragma unroll
  for (int r = 0; r < 8; ++r)
    Cp[(size_t)(mbase + r) * 16 + n] = acc[r];
}

// Y = sum of KSLICES partial products (deterministic split-K combine)
__global__ __launch_bounds__(256) void k_reduce8(const float* __restrict__ Yp,
                                                 float* __restrict__ Y) {
  int idx = blockIdx.x * 256 + threadIdx.x;   // N*16
  float s = 0.f;
#pragma unroll
  for (int p = 0; p < KSLICES; ++p) s += Yp[(size_t)p * N * 16 + idx];
  Y[idx] = s;
}

// Xout = (sigma - deg[i]) * Xin + AX   (i.e. (sigma*I - L) @ X)
__global__ __launch_bounds__(256) void k_bx(const float* __restrict__ Xin,
                                            const float* __restrict__ AX,
                                            const float* __restrict__ deg,
                                            const float* __restrict__ scal,
                                            float* __restrict__ Xout) {
  int idx = blockIdx.x * 256 + threadIdx.x;   // N*16
  int i = idx >> 4;
  float sigma = scal[SC_SIGMA];
  Xout[idx] = (sigma - deg[i]) * Xin[idx] + AX[idx];
}

// LX = deg[i]*X - AX
__global__ __launch_bounds__(256) void k_lx(const float* __restrict__ X,
                                            const float* __restrict__ AX,
                                            const float* __restrict__ deg,
                                            float* __restrict__ LX) {
  int idx = blockIdx.x * 256 + threadIdx.x;
  int i = idx >> 4;
  LX[idx] = deg[i] * X[idx] - AX[idx];
}

// G[a][b] = sum_i P[i][a]*Q[i][b]   (16x16 Gram, one block of 256 threads)
__global__ __launch_bounds__(256) void k_gram(const float* __restrict__ P,
                                              const float* __restrict__ Q,
                                              float* __restrict__ G) {
  int a = threadIdx.x >> 4, b = threadIdx.x & 15;
  float s = 0.f;
  for (int i = 0; i < N; ++i)
    s += P[i * 16 + a] * Q[i * 16 + b];
  G[threadIdx.x] = s;
}

// Cholesky G = R^T R, then Rinv = R^{-1} (upper-tri), serial (16x16 is tiny)
__global__ void k_cholinv(const float* __restrict__ G, float* __restrict__ Rinv) {
  if (threadIdx.x != 0) return;
  float R[16][16], Ri[16][16];
  for (int p = 0; p < 16; ++p)
    for (int q = 0; q < 16; ++q) { R[p][q] = G[p * 16 + q]; Ri[p][q] = 0.f; }
  for (int p = 0; p < 16; ++p) {
    float d = R[p][p];
    for (int k = 0; k < p; ++k) d -= R[k][p] * R[k][p];
    d = sqrtf(fmaxf(d, 1e-20f));
    R[p][p] = d;
    for (int q = p + 1; q < 16; ++q) {
      float s = R[p][q];
      for (int k = 0; k < p; ++k) s -= R[k][p] * R[k][q];
      R[p][q] = s / d;
    }
  }
  for (int p = 15; p >= 0; --p) {
    Ri[p][p] = 1.f / R[p][p];
    for (int q = p + 1; q < 16; ++q) {
      float s = 0.f;
      for (int k = p + 1; k <= q; ++k) s += R[p][k] * Ri[k][q];
      Ri[p][q] = -s / R[p][p];
    }
  }
  for (int p = 0; p < 16; ++p)
    for (int q = 0; q < 16; ++q) Rinv[p * 16 + q] = Ri[p][q];
}

// serial 16x16 Jacobi eigensolver + sort + lambda/done/useF control update
__global__ void k_jacobi(const float* __restrict__ G, float* __restrict__ Vout,
                         float* __restrict__ scal, int initMode) {
  __shared__ float a[16][16], v[16][16], w[16];
  __shared__ int perm[16];
  if (threadIdx.x != 0) return;
  for (int p = 0; p < 16; ++p)
    for (int q = 0; q < 16; ++q) { a[p][q] = G[p * 16 + q]; v[p][q] = (p == q) ? 1.f : 0.f; }
  for (int sweep = 0; sweep < 30; ++sweep) {
    for (int p = 0; p < 15; ++p)
      for (int q = p + 1; q < 16; ++q) {
        float apq = a[p][q];
        if (fabsf(apq) < 1e-14f) continue;
        float theta = 0.5f * (a[q][q] - a[p][p]) / apq;
        float tt = (theta >= 0.f ? 1.f : -1.f) / (fabsf(theta) + sqrtf(theta * theta + 1.f));
        float cc = 1.f / sqrtf(tt * tt + 1.f), ss = tt * cc;
        for (int k = 0; k < 16; ++k) {
          float akp = a[k][p], akq = a[k][q];
          a[k][p] = cc * akp - ss * akq;
          a[k][q] = ss * akp + cc * akq;
        }
        for (int k = 0; k < 16; ++k) {
          float apk = a[p][k], aqk = a[q][k];
          a[p][k] = cc * apk - ss * aqk;
          a[q][k] = ss * apk + cc * aqk;
        }
        for (int k = 0; k < 16; ++k) {
          float vkp = v[k][p], vkq = v[k][q];
          v[k][p] = cc * vkp - ss * vkq;
          v[k][q] = ss * vkp + cc * vkq;
        }
      }
  }
  for (int p = 0; p < 16; ++p) { w[p] = a[p][p]; perm[p] = p; }
  for (int p = 0; p < 15; ++p)
    for (int q = p + 1; q < 16; ++q)
      if (w[perm[q]] < w[perm[p]]) { int tmp = perm[p]; perm[p] = perm[q]; perm[q] = tmp; }
  for (int r = 0; r < 16; ++r)
    for (int k = 0; k < 16; ++k)
      Vout[r * 16 + k] = v[r][perm[k]];
  if (initMode) {
    scal[SC_USEF] = 1.f;
  } else {
    float fn1 = 0.f;
    for (int k = 0; k < CDIM; ++k) fn1 += w[perm[k]];
    float fn2 = fn1 + w[perm[CDIM]];
    int cond1 = fn1 > 1e-10f;
    int cond2 = (!cond1) && (fn2 < 1e-10f);
    if (scal[SC_DONE] == 0.f) {
      float lam = scal[SC_LAM];
      if (cond1) lam *= 2.f; else if (cond2) lam *= 0.5f;
      scal[SC_LAM] = lam;
      scal[SC_USEF] = cond2 ? 0.f : 1.f;
      if (!cond1 && !cond2) scal[SC_DONE] = 1.f;
    } else {
      scal[SC_USEF] = 0.f;
    }
  }
}

__global__ __launch_bounds__(256) void k_fcommit(const float* __restrict__ Fc,
                                                 const float* __restrict__ scal,
                                                 float* __restrict__ F) {
  if (scal[SC_USEF] == 0.f) return;
  int idx = blockIdx.x * 256 + threadIdx.x;
  F[idx] = Fc[idx];
}

// ---------------------------------------------------------------------------
// row-wise simplex projection with fused pairwise distance; one block per row
__global__ __launch_bounds__(256) void k_proj(const float* __restrict__ A0,
                                              const float* __restrict__ U,
                                              const float* __restrict__ F,
                                              const float* __restrict__ scal,
                                              float* __restrict__ Snew) {
  __shared__ float dloc[N];
  __shared__ float uloc[N];   // <0 encodes "masked out"
  __shared__ float redA[256], redB[256];
  __shared__ float fi[CDIM];
  const int i = blockIdx.x, t = threadIdx.x;
  if (t < CDIM) fi[t] = F[i * 16 + t];
  __syncthreads();
  float sqi = 0.f;
#pragma unroll
  for (int c = 0; c < CDIM; ++c) sqi += fi[c] * fi[c];
  const float lamreg = scal[SC_LAM];
  for (int j = t; j < N; j += 256) {
    float av = A0[(size_t)i * N + j];
    float uu = U[(size_t)i * N + j];
    float dot = 0.f, sqj = 0.f;
#pragma unroll
    for (int c = 0; c < CDIM; ++c) {
      float fj = F[j * 16 + c];
      dot += fi[c] * fj; sqj += fj * fj;
    }
    float dist = fmaxf(sqi + sqj - 2.f * dot, 0.f);
    dloc[j] = uu * av - 0.5f * lamreg * dist;
    uloc[j] = (av > 0.f) ? uu : -1.f;
  }
  __syncthreads();
  float mn = 3.4e38f;
  for (int j = t; j < N; j += 256)
    if (uloc[j] > 0.f) mn = fminf(mn, uloc[j] - dloc[j]);
  redA[t] = mn; __syncthreads();
  for (int off = 128; off > 0; off >>= 1) {
    if (t < off) redA[t] = fminf(redA[t], redA[t + off]);
    __syncthreads();
  }
  float lam = redA[0];
  if (!(lam < 3.0e38f)) lam = 0.f;
  __syncthreads();
  for (int it = 0; it < NEWTON; ++it) {
    float fs = 0.f, gs = 0.f;
    for (int j = t; j < N; j += 256) {
      float uu = uloc[j];
      if (uu > 0.f) {
        float v1 = (lam + dloc[j]) / uu;
        if (v1 > 0.f) { fs += v1; gs += 1.f / uu; }
      }
    }
    redA[t] = fs; redB[t] = gs; __syncthreads();
    for (int off = 128; off > 0; off >>= 1) {
      if (t < off) { redA[t] += redA[t + off]; redB[t] += redB[t + off]; }
      __syncthreads();
    }
    float f = redA[0] - 1.f;
    float g = redB[0];
    float step = f / ((g > 0.f) ? g : 1.f);
    if (fabsf(f) > 1e-8f) lam -= step;
    __syncthreads();
  }
  for (int j = t; j < N; j += 256) {
    float uu = uloc[j];
    Snew[(size_t)i * N + j] = (uu > 0.f) ? fmaxf((lam + dloc[j]) / uu, 0.f) : 0.f;
  }
}

// commit (masked by done): S, u, Afin updates
__global__ __launch_bounds__(256) void k_commit(const float* __restrict__ Snew,
                                                const float* __restrict__ A0,
                                                const float* __restrict__ scal,
                                                float* __restrict__ Sout,
                                                float* __restrict__ U,
                                                float* __restrict__ Afin) {
  if (scal[SC_DONE] != 0.f) return;
  size_t idx = (size_t)blockIdx.x * 256 + threadIdx.x;
  int i = (int)(idx >> 10), j = (int)(idx & (N - 1));
  float s  = Snew[idx];
  float st = Snew[(size_t)j * N + i];
  float a0 = A0[idx];
  Sout[idx] = s;
  float diff = s - a0;
  U[idx] = (a0 > 0.f) ? 1.f / (2.f * sqrtf(diff * diff + EPSF)) : 1.f;
  Afin[idx] = 0.5f * (s + st);
}

// ---------------------------------------------------------------------------
__global__ __launch_bounds__(256) void k_labels_init(int* __restrict__ labels) {
  int i = blockIdx.x * 256 + threadIdx.x;
  if (i < N) labels[i] = i;
}

__global__ __launch_bounds__(256) void k_cc_min(const float* __restrict__ Afin,
                                                const int* __restrict__ labels,
                                                int* __restrict__ neigh) {
  __shared__ int red[256];
  int i = blockIdx.x, t = threadIdx.x;
  int mn = labels[i];   // adj includes self
  for (int j = t; j < N; j += 256)
    if (Afin[(size_t)i * N + j] > 0.f) { int lj = labels[j]; mn = (lj < mn) ? lj : mn; }
  red[t] = mn; __syncthreads();
  for (int off = 128; off > 0; off >>= 1) {
    if (t < off) { int o = red[t + off]; if (o < red[t]) red[t] = o; }
    __syncthreads();
  }
  if (t == 0) neigh[i] = red[0];
}

__global__ __launch_bounds__(256) void k_cc_jump(const int* __restrict__ neigh,
                                                 int* __restrict__ labels) {
  int i = blockIdx.x * 256 + threadIdx.x;
  if (i < N) labels[i] = neigh[neigh[i]];
}

__global__ __launch_bounds__(256) void k_write_y(const int* __restrict__ labels,
                                                 float* __restrict__ out) {
  int i = blockIdx.x * 256 + threadIdx.x;
  if (i < N) out[i] = (float)labels[i];
}

// ---------------------------------------------------------------------------
static void big_gemm(const float* A, const float* X, float* Yp, float* Y,
                     hipStream_t stream) {
  k_gemm_wmma<<<dim3(8, KSLICES), 256, 0, stream>>>(A, N, X, Yp, N / KSLICES);
  k_reduce8<<<64, 256, 0, stream>>>(Yp, Y);
}

static void eigh_pipeline(float* Afin, float* deg, float* scal,
                          float* Xa, float* Xb, float* Yt, float* Yp, float* LX,
                          float* G, float* Rinv, float* Vb,
                          float* F, float* Fc, int initMode, hipStream_t stream) {
  k_rowsum<<<N, 256, 0, stream>>>(Afin, deg);
  k_maxdeg<<<1, 256, 0, stream>>>(deg, scal);
  k_initX<<<64, 256, 0, stream>>>(Xa);
  float* cur = Xa; float* alt = Xb;
  for (int t = 0; t < NPOW; ++t) {
    big_gemm(Afin, cur, Yp, Yt, stream);
    k_bx<<<64, 256, 0, stream>>>(cur, Yt, deg, scal, alt);
    float* tmp = cur; cur = alt; alt = tmp;
    if ((t & 3) == 3) {           // Cholesky-QR re-orthonormalization
      k_gram<<<1, 256, 0, stream>>>(cur, cur, G);
      k_cholinv<<<1, 32, 0, stream>>>(G, Rinv);
      k_gemm_wmma<<<dim3(8, 1), 256, 0, stream>>>(cur, 16, Rinv, alt, 16);
      tmp = cur; cur = alt; alt = tmp;
    }
  }
  // Rayleigh-Ritz on the orthonormal subspace
  big_gemm(Afin, cur, Yp, Yt, stream);
  k_lx<<<64, 256, 0, stream>>>(cur, Yt, deg, LX);
  k_gram<<<1, 256, 0, stream>>>(cur, LX, G);
  k_jacobi<<<1, 32, 0, stream>>>(G, Vb, scal, initMode);
  k_gemm_wmma<<<dim3(8, 1), 256, 0, stream>>>(cur, 16, Vb, Fc, 16);
  k_fcommit<<<64, 256, 0, stream>>>(Fc, scal, F);
}

extern "C" void kernel_launch(void* const* d_in, const int* in_sizes, int n_in,
                              void* d_out, int out_size, void* d_ws, size_t ws_size,
                              hipStream_t stream) {
  const float* Ain = (const float*)d_in[0];
  float* out = (float*)d_out;
  float* yOut = out;            // 1024 labels (as float)
  float* Sout = out + N;        // 1024x1024 S

  float* ws = (float*)d_ws;
  size_t off = 0;
  float* A0   = ws + off; off += (size_t)N * N;
  float* Afin = ws + off; off += (size_t)N * N;
  float* U    = ws + off; off += (size_t)N * N;
  float* Snew = ws + off; off += (size_t)N * N;
  float* Xa   = ws + off; off += (size_t)N * 16;
  float* Xb   = ws + off; off += (size_t)N * 16;
  float* Yt   = ws + off; off += (size_t)N * 16;
  float* LX   = ws + off; off += (size_t)N * 16;
  float* F    = ws + off; off += (size_t)N * 16;
  float* Fc   = ws + off; off += (size_t)N * 16;
  float* Yp   = ws + off; off += (size_t)KSLICES * N * 16;
  float* deg  = ws + off; off += N;
  float* G    = ws + off; off += 256;
  float* Rinv = ws + off; off += 256;
  float* Vb   = ws + off; off += 256;
  float* scal = ws + off; off += 16;
  int* labels = (int*)(ws + off); off += N;
  int* neigh  = (int*)(ws + off); off += N;

  k_init<<<1, 32, 0, stream>>>(scal);
  k_preprocess<<<(N * N) / 256, 256, 0, stream>>>(Ain, A0, Afin, U);

  // initial F from L0 = D - (A0+A0^T)/2
  eigh_pipeline(Afin, deg, scal, Xa, Xb, Yt, Yp, LX, G, Rinv, Vb, F, Fc, 1, stream);

  for (int it = 0; it < 6; ++it) {
    k_proj<<<N, 256, 0, stream>>>(A0, U, F, scal, Snew);
    k_commit<<<(N * N) / 256, 256, 0, stream>>>(Snew, A0, scal, Sout, U, Afin);
    eigh_pipeline(Afin, deg, scal, Xa, Xb, Yt, Yp, LX, G, Rinv, Vb, F, Fc, 0, stream);
  }

  k_labels_init<<<4, 256, 0, stream>>>(labels);
  for (int t = 0; t < 16; ++t) {
    k_cc_min<<<N, 256, 0, stream>>>(Afin, labels, neigh);
    k_cc_jump<<<4, 256, 0, stream>>>(neigh, labels);
  }
  k_write_y<<<4, 256, 0, stream>>>(labels, yOut);
}